// GCNGraphClassifier_30090540875859
// MI455X (gfx1250) — compile-verified
//
#include <hip/hip_runtime.h>

#define HID 16
#define NEG_SLOPE 0.2f
#define EPSV 1e-5f
#define NUM_G 1000

typedef float v2f __attribute__((ext_vector_type(2)));
typedef float v8f __attribute__((ext_vector_type(8)));

__device__ __forceinline__ float lrelu(float v) { return v > 0.f ? v : NEG_SLOPE * v; }

// float atomic max via signed-max (v>=0) / unsigned-min (v<0); valid for any
// real float init value, no NaNs in this workload.
__device__ __forceinline__ void atomicMaxF(float* addr, float v) {
  if (v >= 0.f) atomicMax((int*)addr, __float_as_int(v));
  else          atomicMin((unsigned int*)addr, __float_as_uint(v));
}

// ---- padding kernels -------------------------------------------------------
__global__ void k_pad_x(const float* __restrict__ x, float* __restrict__ xp, int N) {
  int t = blockIdx.x * blockDim.x + threadIdx.x;
  if (t >= N * 16) return;
  int n = t >> 4, f = t & 15;
  xp[t] = (f < 9) ? x[n * 9 + f] : 0.f;
}

__global__ void k_pad_w(const float* __restrict__ W, float* __restrict__ Wp) {
  int t = threadIdx.x;            // launched <<<1,256>>>
  int r = t >> 4;
  Wp[t] = (r < 9) ? W[t] : 0.f;   // W1 is (9,16) row-major; same flat index
}

// ---- node GEMM: H[N x 16] = X[N x 16] @ W[16 x 16] via V_WMMA_F32_16X16X4_F32
__global__ void k_gemm16(const float* __restrict__ X, const float* __restrict__ W,
                         float* __restrict__ H, int N) {
  int gid  = blockIdx.x * blockDim.x + threadIdx.x;
  int wave = gid >> 5;
  int lane = threadIdx.x & 31;
  int ntiles = N >> 4;                  // N is a multiple of 16 (500000)
  if (wave >= ntiles) return;           // whole wave exits together -> EXEC all-1s for WMMA
  int base = wave << 4;
  int half = lane >> 4;                 // 0: lanes 0-15, 1: lanes 16-31
  int l16  = lane & 15;
  int row  = base + l16;                // A: lane-major M

  v8f c = {0.f, 0.f, 0.f, 0.f, 0.f, 0.f, 0.f, 0.f};
#pragma unroll
  for (int kc = 0; kc < 4; ++kc) {
    int k0 = kc * 4 + half * 2;         // A 16x4 f32: half-wave holds K offset by 2
    v2f a, b;
    a.x = X[row * 16 + k0];
    a.y = X[row * 16 + k0 + 1];
    b.x = W[(k0)     * 16 + l16];       // B 4x16 f32: row striped across lanes
    b.y = W[(k0 + 1) * 16 + l16];
    c = __builtin_amdgcn_wmma_f32_16x16x4_f32(false, a, false, b, (short)0, c,
                                              false, false);
  }
#pragma unroll
  for (int v = 0; v < 8; ++v)
    H[(size_t)(base + half * 8 + v) * 16 + l16] = c[v];  // C/D: VGPR v = row v (+8)
}

// ---- per-node alpha + self-loop init of segment-max -----------------------
__global__ void k_alpha(const float* __restrict__ H, const float* __restrict__ asrc,
                        const float* __restrict__ adst, float* __restrict__ als,
                        float* __restrict__ ald, float* __restrict__ eself,
                        float* __restrict__ m, int N) {
  int i = blockIdx.x * blockDim.x + threadIdx.x;
  if (i >= N) return;
  float s = 0.f, d = 0.f;
#pragma unroll
  for (int f = 0; f < 16; ++f) {
    float h = H[(size_t)i * 16 + f];
    s += h * asrc[f];
    d += h * adst[f];
  }
  als[i] = s; ald[i] = d;
  float e = lrelu(s + d);               // self-loop edge value
  eself[i] = e;
  m[i] = e;                             // init max with self-loop (always present)
}

__global__ void k_edge_max(const int* __restrict__ src, const int* __restrict__ dst,
                           const float* __restrict__ als, const float* __restrict__ ald,
                           float* __restrict__ m, int E) {
  int e = blockIdx.x * blockDim.x + threadIdx.x;
  if (e >= E) return;
  int s = src[e], d = dst[e];
  atomicMaxF(m + d, lrelu(als[s] + ald[d]));
}

// ---- self-loop contribution initializes ssum and agg (non-atomic) ---------
__global__ void k_self_init(const float* __restrict__ eself, const float* __restrict__ m,
                            const float* __restrict__ H, float* __restrict__ ssum,
                            float* __restrict__ agg, int N) {
  int i = blockIdx.x * blockDim.x + threadIdx.x;
  if (i >= N) return;
  float a0 = expf(eself[i] - m[i]);
  ssum[i] = a0;
  const float4* h4 = (const float4*)(H + (size_t)i * 16);
  float4* o4 = (float4*)(agg + (size_t)i * 16);
#pragma unroll
  for (int v = 0; v < 4; ++v) {
    float4 h = h4[v];
    o4[v] = make_float4(a0 * h.x, a0 * h.y, a0 * h.z, a0 * h.w);
  }
}

// ---- edge softmax weight + aggregation: 16 lanes per edge -----------------
// Lane f==0 of each 16-lane group loads indices + computes the softmax weight,
// then broadcasts (s, d, a) with wave shuffles (ds_bpermute on gfx1250).
__global__ void k_edge_agg(const int* __restrict__ src, const int* __restrict__ dst,
                           const float* __restrict__ als, const float* __restrict__ ald,
                           const float* __restrict__ m, const float* __restrict__ H,
                           float* __restrict__ ssum, float* __restrict__ agg, int E) {
  int t = blockIdx.x * blockDim.x + threadIdx.x;
  int e = t >> 4;
  int f = t & 15;
  int lane = threadIdx.x & 31;
  bool valid = (e < E);

  int s = 0, d = 0;
  float a = 0.f;
  if (valid && f == 0) {
    s = src[e];
    d = dst[e];
    a = expf(lrelu(als[s] + ald[d]) - m[d]);
  }
  // broadcast from lane 0 (group 0) / lane 16 (group 1); executed by all lanes
  int bl = lane & 16;
  s = __shfl(s, bl, 32);
  d = __shfl(d, bl, 32);
  a = __shfl(a, bl, 32);

  if (!valid) return;
  if (f == 0) atomicAdd(ssum + d, a);
  atomicAdd(agg + (size_t)d * 16 + f, a * H[(size_t)s * 16 + f]);
}

// ---- normalize by ssum, add bias -> Y (stored into H buffer) --------------
__global__ void k_y(const float* __restrict__ agg, const float* __restrict__ ssum,
                    const float* __restrict__ b, float* __restrict__ Y, int N) {
  int t = blockIdx.x * blockDim.x + threadIdx.x;
  if (t >= N * 16) return;
  int n = t >> 4, f = t & 15;
  Y[t] = agg[t] / ssum[n] + b[f];
}

// ---- chunked segment-sum (batch is sorted): 16 nodes per thread -----------
__global__ void k_segsum(const float* __restrict__ Y, const int* __restrict__ batch,
                         float* __restrict__ out, int N) {
  int t = blockIdx.x * blockDim.x + threadIdx.x;
  int nch = (N + 15) >> 4;
  if (t >= nch * 16) return;
  int f = t & 15, n0 = (t >> 4) << 4;
  int gcur = -1; float acc = 0.f;
  for (int j = 0; j < 16; ++j) {
    int n = n0 + j;
    if (n >= N) break;
    int g = batch[n];
    if (g != gcur) {
      if (gcur >= 0) atomicAdd(out + (size_t)gcur * 16 + f, acc);
      gcur = g; acc = 0.f;
    }
    acc += Y[(size_t)n * 16 + f];
  }
  if (gcur >= 0) atomicAdd(out + (size_t)gcur * 16 + f, acc);
}

__global__ void k_var(const float* __restrict__ Y, const int* __restrict__ batch,
                      const float* __restrict__ mean, const float* __restrict__ ga,
                      float* __restrict__ var, int N) {
  int t = blockIdx.x * blockDim.x + threadIdx.x;
  int nch = (N + 15) >> 4;
  if (t >= nch * 16) return;
  int f = t & 15, n0 = (t >> 4) << 4;
  float gaf = ga[f];
  int gcur = -1; float acc = 0.f; float mg = 0.f;
  for (int j = 0; j < 16; ++j) {
    int n = n0 + j;
    if (n >= N) break;
    int g = batch[n];
    if (g != gcur) {
      if (gcur >= 0) atomicAdd(var + (size_t)gcur * 16 + f, acc);
      gcur = g; acc = 0.f;
      mg = mean[(size_t)g * 16 + f];
    }
    float sub = Y[(size_t)n * 16 + f] - gaf * mg;
    acc += sub * sub;
  }
  if (gcur >= 0) atomicAdd(var + (size_t)gcur * 16 + f, acc);
}

__global__ void k_div_cnt(float* __restrict__ v, const float* __restrict__ cnt, int G) {
  int t = blockIdx.x * blockDim.x + threadIdx.x;
  if (t >= G * 16) return;
  v[t] /= cnt[t >> 4];
}

__global__ void k_apply(const float* __restrict__ Y, const int* __restrict__ batch,
                        const float* __restrict__ mean, const float* __restrict__ var,
                        const float* __restrict__ gw, const float* __restrict__ gb,
                        const float* __restrict__ ga, const float* __restrict__ xres,
                        float* __restrict__ xout, int N, int has_res) {
  int t = blockIdx.x * blockDim.x + threadIdx.x;
  if (t >= N * 16) return;
  int n = t >> 4, f = t & 15;
  int g = batch[n];
  float sub = Y[t] - ga[f] * mean[(size_t)g * 16 + f];
  float o = gw[f] * sub / sqrtf(var[(size_t)g * 16 + f] + EPSV) + gb[f];
  if (has_res) o += xres[t];
  xout[t] = o > 0.f ? o : 0.f;
}

__global__ void k_count(const int* __restrict__ batch, float* __restrict__ cnt, int N) {
  int i = blockIdx.x * blockDim.x + threadIdx.x;
  if (i >= N) return;
  atomicAdd(cnt + batch[i], 1.0f);
}

__global__ void k_final(const float* __restrict__ pooled, const float* __restrict__ cnt,
                        const float* __restrict__ lw, const float* __restrict__ lb,
                        float* __restrict__ out, int G) {
  int t = blockIdx.x * blockDim.x + threadIdx.x;
  if (t >= G * 2) return;
  int g = t >> 1, c = t & 1;
  float s = 0.f;
#pragma unroll
  for (int f = 0; f < 16; ++f) s += pooled[(size_t)g * 16 + f] * lw[f * 2 + c];
  out[t] = s / cnt[g] + lb[c];
}

static inline int nblk(long long n, int b) { return (int)((n + b - 1) / b); }

extern "C" void kernel_launch(void* const* d_in, const int* in_sizes, int n_in,
                              void* d_out, int out_size, void* d_ws, size_t ws_size,
                              hipStream_t stream) {
  const float* x     = (const float*)d_in[0];
  const int*   ei    = (const int*)d_in[1];
  const int*   batch = (const int*)d_in[2];
  const float* lin_w = (const float*)d_in[32];
  const float* lin_b = (const float*)d_in[33];
  const int N = in_sizes[0] / 9;
  const int E = in_sizes[1] / 2;
  const int G = NUM_G;

  float* ws    = (float*)d_ws;
  float* B0    = ws;
  float* B1    = B0 + (size_t)N * 16;
  float* B2    = B1 + (size_t)N * 16;
  float* als   = B2 + (size_t)N * 16;
  float* ald   = als + N;
  float* eself = ald + N;
  float* mbuf  = eself + N;
  float* ssum  = mbuf + N;
  float* cnt   = ssum + N;        // G
  float* mean  = cnt + G;         // G*16
  float* var   = mean + (size_t)G * 16;  // G*16
  float* wpad  = var + (size_t)G * 16;   // 256

  const int B = 256;

  hipMemsetAsync(cnt, 0, sizeof(float) * G, stream);
  k_pad_x<<<nblk((long long)N * 16, B), B, 0, stream>>>(x, B0, N);
  k_count<<<nblk(N, B), B, 0, stream>>>(batch, cnt, N);

  float* X = B0;
  float* H = B1;
  float* A = B2;

  for (int l = 0; l < 4; ++l) {
    const float* W    = (const float*)d_in[4 + 7 * l];
    const float* asrc = (const float*)d_in[5 + 7 * l];
    const float* adst = (const float*)d_in[6 + 7 * l];
    const float* bb   = (const float*)d_in[7 + 7 * l];
    const float* gw   = (const float*)d_in[8 + 7 * l];
    const float* gb   = (const float*)d_in[9 + 7 * l];
    const float* ga   = (const float*)d_in[10 + 7 * l];

    const float* Wuse = W;
    if (l == 0) {                   // W1 is 9x16 -> zero-pad to 16x16
      k_pad_w<<<1, 256, 0, stream>>>(W, wpad);
      Wuse = wpad;
    }

    int ntiles = N >> 4;
    k_gemm16<<<nblk((long long)ntiles * 32, B), B, 0, stream>>>(X, Wuse, H, N);
    k_alpha<<<nblk(N, B), B, 0, stream>>>(H, asrc, adst, als, ald, eself, mbuf, N);
    k_edge_max<<<nblk(E, B), B, 0, stream>>>(ei, ei + E, als, ald, mbuf, E);
    k_self_init<<<nblk(N, B), B, 0, stream>>>(eself, mbuf, H, ssum, A, N);
    k_edge_agg<<<nblk((long long)E * 16, B), B, 0, stream>>>(ei, ei + E, als, ald,
                                                             mbuf, H, ssum, A, E);
    hipMemsetAsync(mean, 0, sizeof(float) * G * 32, stream);  // mean + var
    k_y<<<nblk((long long)N * 16, B), B, 0, stream>>>(A, ssum, bb, H, N);
    k_segsum<<<nblk(((long long)((N + 15) >> 4)) * 16, B), B, 0, stream>>>(H, batch, mean, N);
    k_div_cnt<<<nblk(G * 16, B), B, 0, stream>>>(mean, cnt, G);
    k_var<<<nblk(((long long)((N + 15) >> 4)) * 16, B), B, 0, stream>>>(H, batch, mean, ga, var, N);
    k_div_cnt<<<nblk(G * 16, B), B, 0, stream>>>(var, cnt, G);
    k_apply<<<nblk((long long)N * 16, B), B, 0, stream>>>(H, batch, mean, var, gw, gb,
                                                          ga, X, A, N, l == 0 ? 0 : 1);
    float* tmp = X; X = A; A = tmp;   // rotate buffers: new X holds layer output
  }

  hipMemsetAsync(mean, 0, sizeof(float) * G * 16, stream);
  k_segsum<<<nblk(((long long)((N + 15) >> 4)) * 16, B), B, 0, stream>>>(X, batch, mean, N);
  k_final<<<nblk(G * 2, B), B, 0, stream>>>(mean, cnt, lin_w, lin_b, (float*)d_out, G);
}